// Attention_21354577395891
// MI455X (gfx1250) — compile-verified
//
#include <hip/hip_runtime.h>

typedef __attribute__((ext_vector_type(2))) float v2f;
typedef __attribute__((ext_vector_type(8))) float v8f;
typedef __attribute__((ext_vector_type(4))) int   v4i;

#if defined(__gfx1250__) && __has_builtin(__builtin_amdgcn_global_load_async_to_lds_b128)
#define USE_ASYNC_LDS 1
#else
#define USE_ASYNC_LDS 0
#endif

namespace {
constexpr int   kB     = 8;
constexpr int   kH     = 32;
constexpr int   kKVH   = 8;
constexpr int   kD     = 128;
constexpr int   kBS    = 16;
constexpr int   kBLKS  = 256;
constexpr int   kWIN   = 1024;
constexpr int   kGQ    = 4;     // H / KVH
constexpr float kScale = 0.08838834764831845f;
constexpr float kNeg   = -1.0e30f;
constexpr int   kLP    = 132;   // LDS row pitch (128 + 4) -> conflict-free strided reads
constexpr int   kWaves = 4;
#if USE_ASYNC_LDS
constexpr int   kNBuf  = 2;     // double-buffered tiles for async pipeline
#else
constexpr int   kNBuf  = 1;
#endif
}

#if USE_ASYNC_LDS
// Async DMA one 16B chunk per lane from global to LDS (no VGPR staging, ASYNCcnt).
// Builtin signature (from clang diagnostic): (int4 AS1* gsrc, int4 AS3* ldst, Ii off, Ii cpol)
__device__ __forceinline__ void async_cp_b128(const float* g, float* l) {
  auto* gp = (__attribute__((address_space(1))) v4i*)((unsigned long long)g);
  auto* lp = (__attribute__((address_space(3))) v4i*)((unsigned)(unsigned long long)l);
  __builtin_amdgcn_global_load_async_to_lds_b128(gp, lp, 0, 0);
}
template <int N>
__device__ __forceinline__ void wait_asynccnt() {
  asm volatile("s_wait_asynccnt %0" :: "n"(N) : "memory");
}
#endif

// lanes 0-15: x[lane]        ; lanes 16-31: y[lane-16]
__device__ __forceinline__ float mix_lo(float x, float y, int lane) {
  float t = __shfl(y, lane & 15, 32);
  return (lane < 16) ? x : t;
}
// lanes 0-15: x[lane+16]     ; lanes 16-31: y[lane]
__device__ __forceinline__ float mix_hi(float x, float y, int lane) {
  float t = __shfl(x, (lane & 15) + 16, 32);
  return (lane < 16) ? t : y;
}

__global__ __launch_bounds__(kWaves * 32, 1)
void paged_attn_decode_wmma(const float* __restrict__ q,
                            const float* __restrict__ knew,
                            const float* __restrict__ vnew,
                            const float* __restrict__ k_cache,
                            const float* __restrict__ v_cache,
                            const float* __restrict__ sinks,
                            const int*   __restrict__ block_tables,
                            const int*   __restrict__ context_lens,
                            float*       __restrict__ out) {
  __shared__ float KT[kWaves][kNBuf][kBS * kLP];  // per-wave K tiles
  __shared__ float VT[kWaves][kNBuf][kBS * kLP];  // per-wave V tiles
  __shared__ float accS[kD * kGQ];                // combined numerator O[d][h]
  __shared__ float smS[kWaves][kGQ];
  __shared__ float slS[kWaves][kGQ];
  __shared__ float msS[kGQ];
  __shared__ float lsumS[kGQ];

  const int b    = blockIdx.x >> 3;   // sequence
  const int g    = blockIdx.x & 7;    // kv head
  const int tid  = threadIdx.x;
  const int wave = tid >> 5;
  const int lane = tid & 31;
  const int hh   = lane & 15;         // col (head) / A row (token or dim) index
  const bool hiHalf = lane >= 16;

  for (int i = tid; i < kD * kGQ; i += kWaves * 32) accS[i] = 0.0f;
  if (tid < kGQ) lsumS[tid] = 0.0f;

  const int ctx      = context_lens[b];
  const int startTok = (ctx > kWIN) ? (ctx - kWIN) : 0;
  const int firstBlk = startTok >> 4;
  const int lastBlk  = (ctx - 1) >> 4;
  const int* bt      = block_tables + b * kBLKS;

  const float* qg  = q    + (size_t)b * kH * kD   + (size_t)g * kGQ * kD;
  const float* kng = knew + (size_t)b * kKVH * kD + (size_t)g * kD;
  const float* vng = vnew + (size_t)b * kKVH * kD + (size_t)g * kD;

  // Precompute Q^T B-fragments for all 32 dim-chunks (heads 4..15 zero padded).
  // B layout (4x16): VGPR0 = rows K=0 (lanes 0-15) / K=2 (lanes 16-31); VGPR1 = K=1 / K=3.
  float qb0[32], qb1[32];
#pragma unroll
  for (int kc = 0; kc < 32; ++kc) {
    const int k0 = kc * 4 + (hiHalf ? 2 : 0);
    float b0 = 0.0f, b1 = 0.0f;
    if (hh < kGQ) {
      b0 = qg[hh * kD + k0];
      b1 = qg[hh * kD + k0 + 1];
    }
    qb0[kc] = b0; qb1[kc] = b1;
  }

  float m_run = kNeg;
  float l_run = 0.0f;
  v8f oacc[8];
#pragma unroll
  for (int t = 0; t < 8; ++t) oacc[t] = {};

  __syncthreads();  // accS/lsumS zero-init visible before combine atomics

  // -------- tile staging (async DMA or batched-register fallback) --------
#if USE_ASYNC_LDS
  auto stage_tile = [&](int tb_, int buf_) {
    const int page = bt[tb_];
    const float* kcb = k_cache + ((size_t)page * kBS * kKVH + g) * kD;
    const float* vcb = v_cache + ((size_t)page * kBS * kKVH + g) * kD;
    float* ktb = &KT[wave][buf_][0];
    float* vtb = &VT[wave][buf_][0];
#pragma unroll
    for (int r = 0; r < kBS; ++r) {
      const int pos = tb_ * kBS + r;
      const float* ks = (pos == ctx - 1) ? kng : (kcb + (size_t)r * kKVH * kD);
      const float* vs = (pos == ctx - 1) ? vng : (vcb + (size_t)r * kKVH * kD);
      async_cp_b128(ks + 4 * lane, ktb + r * kLP + 4 * lane);
      async_cp_b128(vs + 4 * lane, vtb + r * kLP + 4 * lane);
    }
  };
#else
  auto stage_tile = [&](int tb_, int buf_) {
    const int page = bt[tb_];
    const float* kcb = k_cache + ((size_t)page * kBS * kKVH + g) * kD;
    const float* vcb = v_cache + ((size_t)page * kBS * kKVH + g) * kD;
    float* ktb = &KT[wave][buf_][0];
    float* vtb = &VT[wave][buf_][0];
#pragma unroll
    for (int half = 0; half < 2; ++half) {
      float4 kr[8], vr[8];
#pragma unroll
      for (int i = 0; i < 8; ++i) {   // 16 loads in flight
        const int r = half * 8 + i;
        const int pos = tb_ * kBS + r;
        const float* ks = (pos == ctx - 1) ? kng : (kcb + (size_t)r * kKVH * kD);
        const float* vs = (pos == ctx - 1) ? vng : (vcb + (size_t)r * kKVH * kD);
        kr[i] = *(const float4*)(ks + 4 * lane);
        vr[i] = *(const float4*)(vs + 4 * lane);
      }
#pragma unroll
      for (int i = 0; i < 8; ++i) {
        const int r = half * 8 + i;
        *(float4*)(ktb + r * kLP + 4 * lane) = kr[i];
        *(float4*)(vtb + r * kLP + 4 * lane) = vr[i];
      }
    }
  };
#endif

  // -------- per-16-token-block flash step (all WMMA) --------
  auto process_block = [&](int tb_, const float* kt, const float* vt) {
    // S = K_tile x Q^T : C[token][head], K-dim in chunks of 4
    v8f sacc = {};
#pragma unroll
    for (int kc = 0; kc < 32; ++kc) {
      const int k0 = kc * 4 + (hiHalf ? 2 : 0);
      v2f a, bq;
      a[0] = kt[hh * kLP + k0];
      a[1] = kt[hh * kLP + k0 + 1];
      bq[0] = qb0[kc];
      bq[1] = qb1[kc];
      sacc = __builtin_amdgcn_wmma_f32_16x16x4_f32(
          false, a, false, bq, (short)0, sacc, false, false);
    }
    // mask (causal + sliding window) + scale
    float sv[8];
#pragma unroll
    for (int j = 0; j < 8; ++j) {
      const int pos = tb_ * kBS + j + (hiHalf ? 8 : 0);
      const bool valid = (pos >= startTok) && (pos < ctx);
      sv[j] = valid ? sacc[j] * kScale : kNeg;
    }
    // online softmax (per head = per lane column)
    float mb = sv[0];
#pragma unroll
    for (int j = 1; j < 8; ++j) mb = fmaxf(mb, sv[j]);
    mb = fmaxf(mb, __shfl_xor(mb, 16, 32));
    const float m_new = fmaxf(m_run, mb);
    const float alpha = __expf(m_run - m_new);
    float p[8];
    float ls = 0.0f;
#pragma unroll
    for (int j = 0; j < 8; ++j) { p[j] = __expf(sv[j] - m_new); ls += p[j]; }
    ls += __shfl_xor(ls, 16, 32);
    l_run = l_run * alpha + ls;
    m_run = m_new;
#pragma unroll
    for (int t = 0; t < 8; ++t) oacc[t] = oacc[t] * alpha;
    // O^T += V^T x P : A = V (16 dims x 4 tokens), B = P (4 tokens x 16 heads)
#pragma unroll
    for (int c = 0; c < 4; ++c) {
      v2f bp;
      if (c == 0)      { bp[0] = mix_lo(p[0], p[2], lane); bp[1] = mix_lo(p[1], p[3], lane); }
      else if (c == 1) { bp[0] = mix_lo(p[4], p[6], lane); bp[1] = mix_lo(p[5], p[7], lane); }
      else if (c == 2) { bp[0] = mix_hi(p[0], p[2], lane); bp[1] = mix_hi(p[1], p[3], lane); }
      else             { bp[0] = mix_hi(p[4], p[6], lane); bp[1] = mix_hi(p[5], p[7], lane); }
      const int t0 = c * 4 + (hiHalf ? 2 : 0);
#pragma unroll
      for (int tau = 0; tau < 8; ++tau) {
        const int dd = tau * 16 + hh;
        v2f a;
        a[0] = vt[t0 * kLP + dd];
        a[1] = vt[(t0 + 1) * kLP + dd];
        oacc[tau] = __builtin_amdgcn_wmma_f32_16x16x4_f32(
            false, a, false, bp, (short)0, oacc[tau], false, false);
      }
    }
  };

  // -------- main loop over this wave's token blocks --------
  const int tb0 = firstBlk + wave;
#if USE_ASYNC_LDS
  if (tb0 <= lastBlk) stage_tile(tb0, 0);   // prologue DMA
  int cur = 0;
  for (int tb = tb0; tb <= lastBlk; tb += kWaves) {
    const bool hasNext = (tb + kWaves) <= lastBlk;
    if (hasNext) stage_tile(tb + kWaves, cur ^ 1);          // DMA next tile
    if (tb + 2 * kWaves <= lastBlk) {                       // warm L2 two ahead
      const int pg2 = bt[tb + 2 * kWaves];
      const float* base = (hiHalf ? v_cache : k_cache) +
          ((size_t)pg2 * kBS * kKVH + g) * kD + (size_t)(lane & 15) * kKVH * kD;
      __builtin_prefetch(base, 0, 1);
    }
    if (hasNext) wait_asynccnt<32>();       // current tile's 32 copies done (in-order)
    else         wait_asynccnt<0>();
    process_block(tb, &KT[wave][cur][0], &VT[wave][cur][0]);
    cur ^= 1;
  }
#else
  for (int tb = tb0; tb <= lastBlk; tb += kWaves) {
    if (tb + kWaves <= lastBlk) {
      const int pg2 = bt[tb + kWaves];
      const float* base = (hiHalf ? v_cache : k_cache) +
          ((size_t)pg2 * kBS * kKVH + g) * kD + (size_t)(lane & 15) * kKVH * kD;
      __builtin_prefetch(base, 0, 1);
    }
    stage_tile(tb, 0);
    process_block(tb, &KT[wave][0][0], &VT[wave][0][0]);
  }
#endif

  // -------- cross-wave flash combine in LDS --------
  if (lane < kGQ) { smS[wave][lane] = m_run; slS[wave][lane] = l_run; }
  __syncthreads();
  if (tid < kGQ) {
    float ms = smS[0][tid];
#pragma unroll
    for (int w = 1; w < kWaves; ++w) ms = fmaxf(ms, smS[w][tid]);
    msS[tid] = ms;
  }
  __syncthreads();
  float f = 0.0f;
  if (hh < kGQ) f = __expf(m_run - msS[hh]);   // padded heads killed with f=0
  if (lane < kGQ) atomicAdd(&lsumS[lane], l_run * f);
  if (hh < kGQ) {
#pragma unroll
    for (int tau = 0; tau < 8; ++tau) {
#pragma unroll
      for (int j = 0; j < 8; ++j) {
        const int dd = tau * 16 + j + (hiHalf ? 8 : 0);
        atomicAdd(&accS[dd * kGQ + hh], oacc[tau][j] * f);
      }
    }
  }
  __syncthreads();

  // -------- add sink to denominator, normalize, store --------
  for (int idx = tid; idx < kGQ * kD; idx += kWaves * 32) {
    const int h = idx & 3;
    const int d = idx >> 2;
    const float sk    = sinks[g * kGQ + h];
    const float denom = lsumS[h] + __expf(sk - msS[h]);
    out[(size_t)b * kH * kD + (size_t)(g * kGQ + h) * kD + d] = accS[d * kGQ + h] / denom;
  }
}

extern "C" void kernel_launch(void* const* d_in, const int* in_sizes, int n_in,
                              void* d_out, int out_size, void* d_ws, size_t ws_size,
                              hipStream_t stream) {
  (void)in_sizes; (void)n_in; (void)out_size; (void)d_ws; (void)ws_size;
  const float* q     = (const float*)d_in[0];
  const float* k     = (const float*)d_in[1];
  const float* v     = (const float*)d_in[2];
  const float* kc    = (const float*)d_in[3];
  const float* vc    = (const float*)d_in[4];
  const float* sinks = (const float*)d_in[5];
  const int*   bt    = (const int*)d_in[6];
  const int*   cl    = (const int*)d_in[7];
  // d_in[8] (slot_mapping) is implied by block_tables + context_lens; unused.
  dim3 grid(kB * kKVH);        // 64 workgroups: one per (sequence, kv-head)
  dim3 block(kWaves * 32);     // 4 wave32 waves split the attention window
  hipLaunchKernelGGL(paged_attn_decode_wmma, grid, block, 0, stream,
                     q, k, v, kc, vc, sinks, bt, cl, (float*)d_out);
}